// PCTrans_49469433315923
// MI455X (gfx1250) — compile-verified
//
#include <hip/hip_runtime.h>
#include <hip/hip_bf16.h>

typedef float v2f __attribute__((ext_vector_type(2)));
typedef float v8f __attribute__((ext_vector_type(8)));

#define EPS_BN 1e-5f
#define EPS_ATT 1e-9f
#define FLT_BIG 3.402823466e38f

// ---------------------------------------------------------------------------
// helpers
// ---------------------------------------------------------------------------
__device__ __forceinline__ void lds_fence() {
  // wave-private LDS RAW ordering: DS ops are in-order per wave, just drain DScnt
  asm volatile("s_wait_dscnt 0" ::: "memory");
}

__device__ __forceinline__ v8f zero8() {
  v8f z = {0.f, 0.f, 0.f, 0.f, 0.f, 0.f, 0.f, 0.f};
  return z;
}
__device__ __forceinline__ v8f splat8(float x) {
  v8f z = {x, x, x, x, x, x, x, x};
  return z;
}

// D(16x16,f32) += A(16x4,f32) * B(4x16,f32), fp32 WMMA
__device__ __forceinline__ v8f wmma4(float ax, float ay, float bx, float by, v8f c) {
  v2f a, b;
  a.x = ax; a.y = ay;
  b.x = bx; b.y = by;
  return __builtin_amdgcn_wmma_f32_16x16x4_f32(false, a, false, b, (short)0, c,
                                               false, false);
}

// D(16x16) += A(16xK from LDS, lda) * B(Kx16 from LDS, ldb); row==col==lane&15
template <int K>
__device__ __forceinline__ v8f gemm_lds(const float* A, int lda, const float* Bm,
                                        int ldb, v8f c, int row, int hi) {
#pragma unroll
  for (int k = 0; k < K; k += 4) {
    int k0 = k + 2 * hi;
    c = wmma4(A[row * lda + k0], A[row * lda + k0 + 1],
              Bm[k0 * ldb + row], Bm[(k0 + 1) * ldb + row], c);
  }
  return c;
}

__device__ __forceinline__ float loadX(const float* feat, int B, int N, int Cin,
                                       int transposed, int b, int n, int k) {
  if (b >= B || k >= Cin) return 0.f;
  return transposed ? feat[(b * Cin + k) * N + n] : feat[(b * N + n) * Cin + k];
}

// ---------------------------------------------------------------------------
// kernel 0: zero the energy accumulators (atomics -> must reset each launch)
// ---------------------------------------------------------------------------
__global__ void zero_kernel(float* p, int n) {
  int i = blockIdx.x * blockDim.x + threadIdx.x;
  if (i < n) p[i] = 0.f;
}

// ---------------------------------------------------------------------------
// kernel 1: per-CCT energy accumulation.
// wave owns (n, part); for each 16-batch chunk: h = x*linW^T (+b), q = h*qk^T,
// energy[n] += Q^T Q  (rank-4 WMMA updates; A-frag == B-frag by symmetry).
// ---------------------------------------------------------------------------
__global__ void __launch_bounds__(128)
cct_energy_kernel(const float* __restrict__ feat, int B, int N, int Cin,
                  int transposed, const float* __restrict__ lin_w,
                  const float* __restrict__ lin_b, const float* __restrict__ qk_w,
                  float* __restrict__ energy, int nparts) {
  __shared__ float sLin[12 * 64];   // [k][c] = lin_w[c][k], zero-padded K->12
  __shared__ float sQk[64 * 16];    // [k][i] = qk_w[i][k]
  __shared__ float sLinB[64];
  __shared__ float sHt[4][16 * 64];
  __shared__ float sQt[4][16 * 16];

  int tid = threadIdx.x;
  for (int idx = tid; idx < 12 * 64; idx += 128) {
    int k = idx >> 6, c = idx & 63;
    sLin[idx] = (k < Cin) ? lin_w[c * Cin + k] : 0.f;
  }
  for (int idx = tid; idx < 64 * 16; idx += 128) {
    int k = idx >> 4, i = idx & 15;
    sQk[idx] = qk_w[i * 64 + k];
  }
  if (tid < 64) sLinB[tid] = lin_b[tid];
  __syncthreads();

  int w = tid >> 5, lane = tid & 31;
  int wid = blockIdx.x * 4 + w;
  if (wid >= N * nparts) return;
  int n = wid % N, part = wid / N;
  int row = lane & 15, hi = lane >> 4;
  float* ht = sHt[w];
  float* qt = sQt[w];

  int nchunks = (B + 15) >> 4;
  v8f cE = zero8();

  for (int chunk = part; chunk < nchunks; chunk += nparts) {
    int bBase = chunk << 4;
    int b = bBase + row;
    // ---- h = x * lin_w^T : (16x12)*(12x64) -> 4 col tiles ----
    v8f h0 = zero8(), h1 = zero8(), h2 = zero8(), h3 = zero8();
#pragma unroll
    for (int k = 0; k < 12; k += 4) {
      int k0 = k + 2 * hi;
      float ax = loadX(feat, B, N, Cin, transposed, b, n, k0);
      float ay = loadX(feat, B, N, Cin, transposed, b, n, k0 + 1);
      h0 = wmma4(ax, ay, sLin[k0 * 64 + row], sLin[(k0 + 1) * 64 + row], h0);
      h1 = wmma4(ax, ay, sLin[k0 * 64 + 16 + row], sLin[(k0 + 1) * 64 + 16 + row], h1);
      h2 = wmma4(ax, ay, sLin[k0 * 64 + 32 + row], sLin[(k0 + 1) * 64 + 32 + row], h2);
      h3 = wmma4(ax, ay, sLin[k0 * 64 + 48 + row], sLin[(k0 + 1) * 64 + 48 + row], h3);
    }
#pragma unroll
    for (int r = 0; r < 8; r++) {
      int m = r + 8 * hi;
      ht[m * 64 + row]      = h0[r] + sLinB[row];
      ht[m * 64 + 16 + row] = h1[r] + sLinB[16 + row];
      ht[m * 64 + 32 + row] = h2[r] + sLinB[32 + row];
      ht[m * 64 + 48 + row] = h3[r] + sLinB[48 + row];
    }
    lds_fence();
    // ---- q = h * qk^T : (16x64)*(64x16) ----
    v8f cQ = gemm_lds<64>(ht, 64, sQk, 16, zero8(), row, hi);
#pragma unroll
    for (int r = 0; r < 8; r++) qt[(r + 8 * hi) * 16 + row] = cQ[r];
    lds_fence();
    // ---- energy += Q^T Q (symmetric: A frag == B frag) ----
#pragma unroll
    for (int kb = 0; kb < 16; kb += 4) {
      int k0 = kb + 2 * hi;
      float q0 = qt[k0 * 16 + row];
      float q1 = qt[(k0 + 1) * 16 + row];
      cE = wmma4(q0, q1, q0, q1, cE);
    }
  }
#pragma unroll
  for (int r = 0; r < 8; r++)
    atomicAdd(&energy[n * 256 + (r + 8 * hi) * 16 + row], cE[r]);
}

// ---------------------------------------------------------------------------
// kernel 2: att = softmax(energy, axis=-1); att /= (eps + colsum over axis=1)
// one block per n, 256 threads = one (i,j) each.
// ---------------------------------------------------------------------------
__global__ void __launch_bounds__(256)
att_kernel(const float* __restrict__ energy, float* __restrict__ att) {
  __shared__ float sE[256];
  __shared__ float rowred[16];
  __shared__ float colsum[16];
  int n = blockIdx.x, tid = threadIdx.x;
  int i = tid >> 4, j = tid & 15;
  float e = energy[n * 256 + tid];
  sE[tid] = e;
  __syncthreads();
  if (j == 0) {
    float m = -FLT_BIG;
    for (int jj = 0; jj < 16; jj++) m = fmaxf(m, sE[i * 16 + jj]);
    rowred[i] = m;
  }
  __syncthreads();
  float ex = __expf(e - rowred[i]);
  __syncthreads();
  sE[tid] = ex;
  __syncthreads();
  if (j == 0) {
    float s = 0.f;
    for (int jj = 0; jj < 16; jj++) s += sE[i * 16 + jj];
    rowred[i] = s;
  }
  __syncthreads();
  float a = ex / rowred[i];
  sE[tid] = a;
  __syncthreads();
  if (i == 0) {
    float s = 0.f;
    for (int ii = 0; ii < 16; ii++) s += sE[ii * 16 + j];
    colsum[j] = s;
  }
  __syncthreads();
  att[n * 256 + tid] = a / (EPS_ATT + colsum[j]);
}

// ---------------------------------------------------------------------------
// kernel 3: per-CCT main pass. wave owns a 16-batch chunk; loops n:
// h -> v -> x_r = v*att[n] -> tr = x_r*t^T -> BN/ReLU -> running max over n.
// ---------------------------------------------------------------------------
__global__ void __launch_bounds__(64)
cct_main_kernel(const float* __restrict__ feat, int B, int N, int Cin,
                int transposed, const float* __restrict__ lin_w,
                const float* __restrict__ lin_b, const float* __restrict__ v_w,
                const float* __restrict__ v_b, const float* __restrict__ t_w,
                const float* __restrict__ t_b, const float* __restrict__ bn_g,
                const float* __restrict__ bn_b, const float* __restrict__ bn_m,
                const float* __restrict__ bn_v, const float* __restrict__ att,
                float* __restrict__ fout) {
  __shared__ float sLin[12 * 64];
  __shared__ float sVw[64 * 16];  // [k][i] = v_w[i][k]
  __shared__ float sTw[16 * 64];  // [k][c] = t_w[c][k]
  __shared__ float sAtt[32 * 256];
  __shared__ float sLinB[64], sTb[64], sScale[64], sShift[64], sVb[16];
  __shared__ float sHt[2][16 * 64];
  __shared__ float sVt[2][16 * 16];
  __shared__ float sXr[2][16 * 16];

  int tid = threadIdx.x;
  for (int idx = tid; idx < 12 * 64; idx += 64) {
    int k = idx >> 6, c = idx & 63;
    sLin[idx] = (k < Cin) ? lin_w[c * Cin + k] : 0.f;
  }
  for (int idx = tid; idx < 64 * 16; idx += 64) {
    int k = idx >> 4, i = idx & 15;
    sVw[idx] = v_w[i * 64 + k];
  }
  for (int idx = tid; idx < 16 * 64; idx += 64) {
    int k = idx >> 6, c = idx & 63;
    sTw[idx] = t_w[c * 16 + k];
  }
  for (int idx = tid; idx < N * 256; idx += 64) sAtt[idx] = att[idx];
  if (tid < 64) {
    sLinB[tid] = lin_b[tid];
    sTb[tid] = t_b[tid];
    float sc = bn_g[tid] * rsqrtf(bn_v[tid] + EPS_BN);
    sScale[tid] = sc;
    sShift[tid] = bn_b[tid] - bn_m[tid] * sc;
  }
  if (tid < 16) sVb[tid] = v_b[tid];
  __syncthreads();

  int w = tid >> 5, lane = tid & 31;
  int nchunks = (B + 15) >> 4;
  int chunk = blockIdx.x * 2 + w;
  if (chunk >= nchunks) return;
  int row = lane & 15, hi = lane >> 4;
  int bBase = chunk << 4;
  int b = bBase + row;
  float* ht = sHt[w];
  float* vt = sVt[w];
  float* xrt = sXr[w];

  if (b < B)  // warm L2/WGP$ for this chunk's features
    __builtin_prefetch(feat + (transposed ? (size_t)b * Cin * N : (size_t)b * N * Cin), 0, 3);

  v8f run0 = splat8(-FLT_BIG), run1 = splat8(-FLT_BIG);
  v8f run2 = splat8(-FLT_BIG), run3 = splat8(-FLT_BIG);

  for (int n = 0; n < N; n++) {
    // ---- h (biased) ----
    v8f h0 = zero8(), h1 = zero8(), h2 = zero8(), h3 = zero8();
#pragma unroll
    for (int k = 0; k < 12; k += 4) {
      int k0 = k + 2 * hi;
      float ax = loadX(feat, B, N, Cin, transposed, b, n, k0);
      float ay = loadX(feat, B, N, Cin, transposed, b, n, k0 + 1);
      h0 = wmma4(ax, ay, sLin[k0 * 64 + row], sLin[(k0 + 1) * 64 + row], h0);
      h1 = wmma4(ax, ay, sLin[k0 * 64 + 16 + row], sLin[(k0 + 1) * 64 + 16 + row], h1);
      h2 = wmma4(ax, ay, sLin[k0 * 64 + 32 + row], sLin[(k0 + 1) * 64 + 32 + row], h2);
      h3 = wmma4(ax, ay, sLin[k0 * 64 + 48 + row], sLin[(k0 + 1) * 64 + 48 + row], h3);
    }
#pragma unroll
    for (int r = 0; r < 8; r++) {
      int m = r + 8 * hi;
      h0[r] += sLinB[row];
      h1[r] += sLinB[16 + row];
      h2[r] += sLinB[32 + row];
      h3[r] += sLinB[48 + row];
      ht[m * 64 + row]      = h0[r];
      ht[m * 64 + 16 + row] = h1[r];
      ht[m * 64 + 32 + row] = h2[r];
      ht[m * 64 + 48 + row] = h3[r];
    }
    lds_fence();
    // ---- v = h * v_w^T + v_b ----
    v8f cV = gemm_lds<64>(ht, 64, sVw, 16, zero8(), row, hi);
#pragma unroll
    for (int r = 0; r < 8; r++) vt[(r + 8 * hi) * 16 + row] = cV[r] + sVb[row];
    lds_fence();
    // ---- x_r = v * att[n] ----
    v8f cX = gemm_lds<16>(vt, 16, &sAtt[n * 256], 16, zero8(), row, hi);
#pragma unroll
    for (int r = 0; r < 8; r++) xrt[(r + 8 * hi) * 16 + row] = cX[r];
    lds_fence();
    // ---- tr = x_r * t_w^T ----
    v8f t0 = gemm_lds<16>(xrt, 16, sTw, 64, zero8(), row, hi);
    v8f t1 = gemm_lds<16>(xrt, 16, sTw + 16, 64, zero8(), row, hi);
    v8f t2 = gemm_lds<16>(xrt, 16, sTw + 32, 64, zero8(), row, hi);
    v8f t3 = gemm_lds<16>(xrt, 16, sTw + 48, 64, zero8(), row, hi);
    // ---- BN / ReLU / running max ----
#pragma unroll
    for (int r = 0; r < 8; r++) {
      float hh, tr, bn;
      hh = h0[r]; tr = t0[r] + sTb[row];
      bn = (hh - tr) * sScale[row] + sShift[row];
      run0[r] = fmaxf(run0[r], hh + fmaxf(bn, 0.f));
      hh = h1[r]; tr = t1[r] + sTb[16 + row];
      bn = (hh - tr) * sScale[16 + row] + sShift[16 + row];
      run1[r] = fmaxf(run1[r], hh + fmaxf(bn, 0.f));
      hh = h2[r]; tr = t2[r] + sTb[32 + row];
      bn = (hh - tr) * sScale[32 + row] + sShift[32 + row];
      run2[r] = fmaxf(run2[r], hh + fmaxf(bn, 0.f));
      hh = h3[r]; tr = t3[r] + sTb[48 + row];
      bn = (hh - tr) * sScale[48 + row] + sShift[48 + row];
      run3[r] = fmaxf(run3[r], hh + fmaxf(bn, 0.f));
    }
  }
#pragma unroll
  for (int r = 0; r < 8; r++) {
    int bb = bBase + r + 8 * hi;
    if (bb < B) {
      fout[bb * 64 + row]      = run0[r];
      fout[bb * 64 + 16 + row] = run1[r];
      fout[bb * 64 + 32 + row] = run2[r];
      fout[bb * 64 + 48 + row] = run3[r];
    }
  }
}

// ---------------------------------------------------------------------------
// kernel 4: PCT head, one block (64 threads) per batch element.
// ---------------------------------------------------------------------------
__global__ void __launch_bounds__(64)
pct_kernel(const float* __restrict__ f1, const float* __restrict__ f2,
           const float* __restrict__ f12, const float* __restrict__ t_w,
           const float* __restrict__ t_b, const float* __restrict__ g,
           const float* __restrict__ bb, const float* __restrict__ mm,
           const float* __restrict__ vv, const float* __restrict__ fc_w,
           const float* __restrict__ fc_b, float* __restrict__ out, int B) {
  __shared__ float sx1[64], sx2[64], sx12[64], sxr[64], scol[64];
  __shared__ float sE[64 * 64];
  int b = blockIdx.x;
  int i = threadIdx.x;
  sx1[i] = f1[b * 64 + i];
  sx2[i] = f2[b * 64 + i];
  sx12[i] = f12[b * 64 + i];
  __syncthreads();
  float x1i = sx1[i];
  // softmax over j of x1[i]*x2[j]
  float mx = -FLT_BIG;
  for (int j = 0; j < 64; j++) mx = fmaxf(mx, x1i * sx2[j]);
  float s = 0.f;
  for (int j = 0; j < 64; j++) {
    float ex = __expf(x1i * sx2[j] - mx);
    sE[i * 64 + j] = ex;
    s += ex;
  }
  float inv = 1.f / s;
  for (int j = 0; j < 64; j++) sE[i * 64 + j] *= inv;
  __syncthreads();
  // column sums (thread i handles column j=i)
  float cs = 0.f;
  for (int k = 0; k < 64; k++) cs += sE[k * 64 + i];
  scol[i] = cs;
  __syncthreads();
  // x_r[i] = sum_j e[i][j]/(eps+colsum[j]) * x12[j]
  float xr = 0.f;
  for (int j = 0; j < 64; j++)
    xr += sE[i * 64 + j] / (EPS_ATT + scol[j]) * sx12[j];
  sxr[i] = xr;
  __syncthreads();
  // tr, BN, ReLU
  float tr = t_b[i];
  for (int k = 0; k < 64; k++) tr += t_w[i * 64 + k] * sxr[k];
  float sc = g[i] * rsqrtf(vv[i] + EPS_BN);
  float bn = (x1i - tr - mm[i]) * sc + bb[i];
  float relu = fmaxf(bn, 0.f);
  // gate = sigmoid([e, e^T, x1] . fc_w + fc_b)
  float dot = fc_b[0] + x1i * fc_w[128];
  for (int j = 0; j < 64; j++)
    dot += sE[i * 64 + j] * fc_w[j] + sE[j * 64 + i] * fc_w[64 + j];
  float gate = 1.f / (1.f + __expf(-dot));
  out[b * 64 + i] = gate * x1i + relu;
}

// ---------------------------------------------------------------------------
// launch
// ---------------------------------------------------------------------------
extern "C" void kernel_launch(void* const* d_in, const int* in_sizes, int n_in,
                              void* d_out, int out_size, void* d_ws,
                              size_t ws_size, hipStream_t stream) {
  const float* feat = (const float*)d_in[0];
  int B = in_sizes[0] / (32 * 11);
  auto P = [&](int i) { return (const float*)d_in[i]; };

  float* w = (float*)d_ws;
  float* e1 = w;
  float* e12 = w + 8192;
  float* e2 = w + 16384;
  float* a1 = w + 24576;
  float* a12 = w + 32768;
  float* a2 = w + 40960;
  float* f1 = w + 49152;
  float* f12 = f1 + (size_t)B * 64;
  float* f2 = f12 + (size_t)B * 64;

  zero_kernel<<<(24576 + 255) / 256, 256, 0, stream>>>(w, 24576);

  const int NPARTS = 125;
  int nchunks = (B + 15) / 16;

  {
    int grid1 = (32 * NPARTS + 3) / 4;
    cct_energy_kernel<<<grid1, 128, 0, stream>>>(feat, B, 32, 11, 0, P(1), P(2),
                                                 P(3), e1, NPARTS);
    cct_energy_kernel<<<grid1, 128, 0, stream>>>(feat, B, 32, 11, 0, P(12),
                                                 P(13), P(14), e12, NPARTS);
    int grid2 = (11 * NPARTS + 3) / 4;
    cct_energy_kernel<<<grid2, 128, 0, stream>>>(feat, B, 11, 32, 1, P(23),
                                                 P(24), P(25), e2, NPARTS);
  }
  att_kernel<<<32, 256, 0, stream>>>(e1, a1);
  att_kernel<<<32, 256, 0, stream>>>(e12, a12);
  att_kernel<<<11, 256, 0, stream>>>(e2, a2);

  int gmain = (nchunks + 1) / 2;
  cct_main_kernel<<<gmain, 64, 0, stream>>>(feat, B, 32, 11, 0, P(1), P(2),
                                            P(4), P(5), P(6), P(7), P(8), P(9),
                                            P(10), P(11), a1, f1);
  cct_main_kernel<<<gmain, 64, 0, stream>>>(feat, B, 32, 11, 0, P(12), P(13),
                                            P(15), P(16), P(17), P(18), P(19),
                                            P(20), P(21), P(22), a12, f12);
  cct_main_kernel<<<gmain, 64, 0, stream>>>(feat, B, 11, 32, 1, P(23), P(24),
                                            P(26), P(27), P(28), P(29), P(30),
                                            P(31), P(32), P(33), a2, f2);

  pct_kernel<<<B, 64, 0, stream>>>(f1, f2, f12, P(34), P(35), P(36), P(37),
                                   P(38), P(39), P(40), P(41), (float*)d_out, B);
}